// GCN_83777632075847
// MI455X (gfx1250) — compile-verified
//
#include <hip/hip_runtime.h>

typedef __attribute__((ext_vector_type(2))) float v2f;
typedef __attribute__((ext_vector_type(8))) float v8f;

#define N_NODES   100000
#define N_EDGES   1600000
#define IN_FEAT   128
#define HIDDEN    128
#define N_CLASSES 40

// ---------------- degree / normalization ----------------

__global__ __launch_bounds__(256) void k_deg_init(float* __restrict__ deg) {
  int i = blockIdx.x * 256 + threadIdx.x;
  if (i < N_NODES) deg[i] = 1.0f;  // self-loop
}

__global__ __launch_bounds__(256) void k_deg_edges(const long long* __restrict__ col,
                                                   float* __restrict__ deg) {
  int e = blockIdx.x * 256 + threadIdx.x;
  if (e < N_EDGES) atomicAdd(&deg[(int)col[e]], 1.0f);
}

__global__ __launch_bounds__(256) void k_rsqrt(float* __restrict__ deg) {
  int i = blockIdx.x * 256 + threadIdx.x;
  if (i < N_NODES) deg[i] = rsqrtf(deg[i]);
}

// ---------------- GEMM1: h1 = x @ W1   (fp32 WMMA 16x16x4) ----------------
// One wave computes one 16(M) x 16(N) tile; 8 waves/block cover N=128.
// Epilogue fuses the self-loop init: a1 = dis^2 * h1 (saves a 102MB pass).

__global__ __launch_bounds__(256) void k_gemm1(const float* __restrict__ X,
                                               const float* __restrict__ W,
                                               const float* __restrict__ dis,
                                               float* __restrict__ H,
                                               float* __restrict__ A1) {
  const int lane = threadIdx.x & 31;
  const int wave = threadIdx.x >> 5;      // n-tile 0..7
  const int m16  = lane & 15;
  const int kh   = lane >> 4;             // 0 or 1
  const int rowBase = blockIdx.x * 16;
  const int nG = wave * 16 + m16;

  const float* xrow = X + (size_t)(rowBase + m16) * IN_FEAT + kh * 2;
  const float* wcol = W + nG + kh * 2 * HIDDEN;

  v8f acc = {};
#pragma unroll
  for (int k0 = 0; k0 < IN_FEAT; k0 += 4) {
    v2f a, b;
    float2 av = *(const float2*)(xrow + k0);           // contiguous 8B load
    a.x = av.x; a.y = av.y;
    b.x = wcol[(size_t)k0 * HIDDEN];
    b.y = wcol[(size_t)k0 * HIDDEN + HIDDEN];
    acc = __builtin_amdgcn_wmma_f32_16x16x4_f32(false, a, false, b,
                                                (short)0, acc, false, false);
  }
#pragma unroll
  for (int r = 0; r < 8; ++r) {
    int m = rowBase + r + 8 * kh;
    float v = acc[r];
    float d = dis[m];
    H [(size_t)m * HIDDEN + nG] = v;
    A1[(size_t)m * HIDDEN + nG] = d * d * v;   // self-loop contribution
  }
}

// ---------------- edge scatter (layer 1, F=128): one wave per edge ----------------

__global__ __launch_bounds__(256) void k_scatter1(const long long* __restrict__ ei,
                                                  const float* __restrict__ dis,
                                                  const float* __restrict__ H,
                                                  float* __restrict__ A1) {
  const int lane = threadIdx.x & 31;
  const int e = blockIdx.x * 8 + (threadIdx.x >> 5);
  if (e >= N_EDGES) return;
  const int r = (int)ei[e];
  const int c = (int)ei[(size_t)N_EDGES + e];
  const float norm = dis[r] * dis[c];
  const float4 hv = *(const float4*)(H + (size_t)r * HIDDEN + lane * 4);
  float* dst = A1 + (size_t)c * HIDDEN + lane * 4;
  atomicAdd(dst + 0, norm * hv.x);
  atomicAdd(dst + 1, norm * hv.y);
  atomicAdd(dst + 2, norm * hv.z);
  atomicAdd(dst + 3, norm * hv.w);
}

// ---------------- GEMM2: h2 = relu(a1 + b1) @ W2   (fp32 WMMA, N padded to 48) ----
// Epilogue fuses self-loop init of the output: OUT = dis^2 * h2.

__global__ __launch_bounds__(96) void k_gemm2(const float* __restrict__ A1,
                                              const float* __restrict__ B1,
                                              const float* __restrict__ W2,
                                              const float* __restrict__ dis,
                                              float* __restrict__ H2,
                                              float* __restrict__ OUT) {
  const int lane = threadIdx.x & 31;
  const int wave = threadIdx.x >> 5;      // n-tile 0..2
  const int m16  = lane & 15;
  const int kh   = lane >> 4;
  const int rowBase = blockIdx.x * 16;
  const int nG = wave * 16 + m16;         // 0..47
  const bool nOK = nG < N_CLASSES;

  const float* arow = A1 + (size_t)(rowBase + m16) * HIDDEN + kh * 2;

  v8f acc = {};
#pragma unroll
  for (int k0 = 0; k0 < HIDDEN; k0 += 4) {
    int k = k0 + kh * 2;
    v2f a, b;
    float2 av = *(const float2*)(arow + k0);
    a.x = fmaxf(av.x + B1[k], 0.0f);      // fused bias + ReLU
    a.y = fmaxf(av.y + B1[k + 1], 0.0f);
    b.x = nOK ? W2[(size_t)k * N_CLASSES + nG] : 0.0f;
    b.y = nOK ? W2[(size_t)(k + 1) * N_CLASSES + nG] : 0.0f;
    acc = __builtin_amdgcn_wmma_f32_16x16x4_f32(false, a, false, b,
                                                (short)0, acc, false, false);
  }
  if (nOK) {
#pragma unroll
    for (int r = 0; r < 8; ++r) {
      int m = rowBase + r + 8 * kh;
      float v = acc[r];
      float d = dis[m];
      H2 [(size_t)m * N_CLASSES + nG] = v;
      OUT[(size_t)m * N_CLASSES + nG] = d * d * v;  // self-loop contribution
    }
  }
}

// ---------------- edge scatter (layer 2, F=40): one wave per edge ----------------

__global__ __launch_bounds__(256) void k_scatter2(const long long* __restrict__ ei,
                                                  const float* __restrict__ dis,
                                                  const float* __restrict__ H2,
                                                  float* __restrict__ OUT) {
  const int lane = threadIdx.x & 31;
  const int e = blockIdx.x * 8 + (threadIdx.x >> 5);
  if (e >= N_EDGES) return;
  const int r = (int)ei[e];
  const int c = (int)ei[(size_t)N_EDGES + e];
  const float norm = dis[r] * dis[c];
  const float* src = H2 + (size_t)r * N_CLASSES;
  float* dst = OUT + (size_t)c * N_CLASSES;
  atomicAdd(dst + lane, norm * src[lane]);
  if (lane < N_CLASSES - 32)
    atomicAdd(dst + 32 + lane, norm * src[32 + lane]);
}

// ---------------- bias + log_softmax, one wave32 per row ----------------

__global__ __launch_bounds__(256) void k_logsoftmax(float* __restrict__ OUT,
                                                    const float* __restrict__ b2) {
  const int lane = threadIdx.x & 31;
  const int row = blockIdx.x * 8 + (threadIdx.x >> 5);
  if (row >= N_NODES) return;
  float* p = OUT + (size_t)row * N_CLASSES;
  const bool has2 = lane < (N_CLASSES - 32);
  float v0 = p[lane] + b2[lane];
  float v1 = has2 ? (p[lane + 32] + b2[lane + 32]) : -3.4028235e38f;
  float m = fmaxf(v0, v1);
#pragma unroll
  for (int off = 16; off; off >>= 1) m = fmaxf(m, __shfl_xor(m, off, 32));
  float s = expf(v0 - m) + (has2 ? expf(v1 - m) : 0.0f);
#pragma unroll
  for (int off = 16; off; off >>= 1) s += __shfl_xor(s, off, 32);
  float lse = m + logf(s);
  p[lane] = v0 - lse;
  if (has2) p[lane + 32] = v1 - lse;
}

// ---------------- launch ----------------

extern "C" void kernel_launch(void* const* d_in, const int* in_sizes, int n_in,
                              void* d_out, int out_size, void* d_ws, size_t ws_size,
                              hipStream_t stream) {
  const float*     x  = (const float*)d_in[0];
  const long long* ei = (const long long*)d_in[1];   // int64 edge_index [2, E]
  const float*     W1 = (const float*)d_in[2];
  const float*     b1 = (const float*)d_in[3];
  const float*     W2 = (const float*)d_in[4];
  const float*     b2 = (const float*)d_in[5];
  float* out = (float*)d_out;

  // carve workspace (256B aligned regions)
  char* ws = (char*)d_ws;
  size_t off = 0;
  auto carve = [&](size_t bytes) -> void* {
    off = (off + 255) & ~(size_t)255;
    void* p = ws + off;
    off += bytes;
    return p;
  };
  float* h1  = (float*)carve(sizeof(float) * (size_t)N_NODES * HIDDEN);     // 51.2 MB
  float* a1  = (float*)carve(sizeof(float) * (size_t)N_NODES * HIDDEN);     // 51.2 MB
  float* h2  = (float*)carve(sizeof(float) * (size_t)N_NODES * N_CLASSES);  // 16.0 MB
  float* dis = (float*)carve(sizeof(float) * (size_t)N_NODES);              //  0.4 MB
  (void)ws_size; (void)in_sizes; (void)n_in; (void)out_size;

  // symmetric degree normalization (shared by both layers)
  k_deg_init <<<(N_NODES + 255) / 256, 256, 0, stream>>>(dis);
  k_deg_edges<<<(N_EDGES + 255) / 256, 256, 0, stream>>>(ei + N_EDGES, dis);
  k_rsqrt    <<<(N_NODES + 255) / 256, 256, 0, stream>>>(dis);

  // layer 1: GEMM (+ fused self-loop init) then edge scatter
  k_gemm1   <<<N_NODES / 16, 256, 0, stream>>>(x, W1, dis, h1, a1);
  k_scatter1<<<N_EDGES / 8, 256, 0, stream>>>(ei, dis, h1, a1);

  // layer 2: GEMM (bias1+ReLU fused in, self-loop init fused out) then scatter
  k_gemm2   <<<N_NODES / 16, 96, 0, stream>>>(a1, b1, W2, dis, h2, out);
  k_scatter2<<<N_EDGES / 8, 256, 0, stream>>>(ei, dis, h2, out);

  // bias2 + log_softmax
  k_logsoftmax<<<(N_NODES + 7) / 8, 256, 0, stream>>>(out, b2);
}